// SelfAttention_72816875536586
// MI455X (gfx1250) — compile-verified
//
#include <hip/hip_runtime.h>
#include <hip/hip_bf16.h>

typedef float v2f __attribute__((ext_vector_type(2)));
typedef float v8f __attribute__((ext_vector_type(8)));

#define D_DIM 1024
#define S_DIM 4096
#define B_DIM 4

// ---------------------------------------------------------------------------
// Wave-level 32(M) x 64(N) x K fp32 GEMM tile using V_WMMA_F32_16X16X4_F32.
// aptr: &A[(m0 + lane%16) * lda + 2*(lane/16)]   (A row-major, K contiguous)
// bptr: &B[(n0 + lane%16) * ldb + 2*(lane/16)]   (B^T row-major, K contiguous)
// a_frag_stride = 16*lda, b_frag_stride = 16*ldb (element strides)
// ---------------------------------------------------------------------------
__device__ __forceinline__ void wave_gemm_32x64(const float* __restrict__ aptr,
                                                long long a_frag_stride,
                                                const float* __restrict__ bptr,
                                                long long b_frag_stride,
                                                int K, v8f acc[2][4]) {
  for (int k = 0; k < K; k += 4) {
    v2f a0 = *(const v2f*)(aptr + k);
    v2f a1 = *(const v2f*)(aptr + a_frag_stride + k);
#pragma unroll
    for (int j = 0; j < 4; ++j) {
      v2f b = *(const v2f*)(bptr + j * b_frag_stride + k);
      acc[0][j] = __builtin_amdgcn_wmma_f32_16x16x4_f32(
          false, a0, false, b, (short)0, acc[0][j], false, false);
      acc[1][j] = __builtin_amdgcn_wmma_f32_16x16x4_f32(
          false, a1, false, b, (short)0, acc[1][j], false, false);
    }
  }
}

// ---------------------------------------------------------------------------
// Kernel 1: q/k/v = batch @ W.T + b.  blockIdx.z selects Wq/Wk/Wv.
// q and k are stored TRANSPOSED per batch: qT[b][d][s], kT[b][d][s].
// v is stored natural: v[b][s][d].
// grid: (64, 16, 3), block: 256 (8 waves of 32); wave tile 32x64.
// ---------------------------------------------------------------------------
__global__ void proj_qkv_kernel(const float* __restrict__ batch,
                                const float* __restrict__ Wq, const float* __restrict__ bq,
                                const float* __restrict__ Wk, const float* __restrict__ bk,
                                const float* __restrict__ Wv, const float* __restrict__ bv,
                                float* __restrict__ qT, float* __restrict__ kT,
                                float* __restrict__ vv) {
  const int lane = threadIdx.x & 31;
  const int wave = threadIdx.x >> 5;
  const int lr = lane & 15;
  const int hi = lane >> 4;
  const int m0 = blockIdx.x * 256 + wave * 32;   // row in flattened [B*S]
  const int n0 = blockIdx.y * 64;                // output feature
  const int which = blockIdx.z;

  const float* W    = (which == 0) ? Wq : (which == 1) ? Wk : Wv;
  const float* bias = (which == 0) ? bq : (which == 1) ? bk : bv;

  const float* aptr = batch + (long long)(m0 + lr) * D_DIM + 2 * hi;
  const float* bptr = W     + (long long)(n0 + lr) * D_DIM + 2 * hi;

  v8f acc[2][4] = {};
  wave_gemm_32x64(aptr, 16LL * D_DIM, bptr, 16LL * D_DIM, D_DIM, acc);

  if (which < 2) {
    float* dstT = (which == 0) ? qT : kT;
#pragma unroll
    for (int i = 0; i < 2; ++i)
#pragma unroll
      for (int j = 0; j < 4; ++j) {
        const int n = n0 + 16 * j + lr;
        const float bn = bias[n];
#pragma unroll
        for (int r = 0; r < 8; ++r) {
          const int m = m0 + 16 * i + r + 8 * hi;     // global [B*S] row
          const int bb = m >> 12;                     // /4096
          const int s  = m & (S_DIM - 1);
          dstT[((long long)bb * D_DIM + n) * S_DIM + s] = acc[i][j][r] + bn;
        }
      }
  } else {
#pragma unroll
    for (int i = 0; i < 2; ++i)
#pragma unroll
      for (int j = 0; j < 4; ++j) {
        const int n = n0 + 16 * j + lr;
        const float bn = bias[n];
#pragma unroll
        for (int r = 0; r < 8; ++r) {
          const int m = m0 + 16 * i + r + 8 * hi;
          vv[(long long)m * D_DIM + n] = acc[i][j][r] + bn;
        }
      }
  }
}

// ---------------------------------------------------------------------------
// Kernel 2: energy[b][d][e] = sum_s qT[b][d][s] * kT[b][e][s]
// M=N=1024, K=4096 per batch.  grid: (4, 16, 4), block 256.
// Output written directly into the attn region of d_out (softmaxed in place).
// ---------------------------------------------------------------------------
__global__ void energy_kernel(const float* __restrict__ qT,
                              const float* __restrict__ kT,
                              float* __restrict__ energy) {
  const int lane = threadIdx.x & 31;
  const int wave = threadIdx.x >> 5;
  const int lr = lane & 15;
  const int hi = lane >> 4;
  const int m0 = blockIdx.x * 256 + wave * 32;   // d
  const int n0 = blockIdx.y * 64;                // e
  const int b  = blockIdx.z;

  const float* qb = qT + (long long)b * D_DIM * S_DIM;
  const float* kb = kT + (long long)b * D_DIM * S_DIM;
  const float* aptr = qb + (long long)(m0 + lr) * S_DIM + 2 * hi;
  const float* bptr = kb + (long long)(n0 + lr) * S_DIM + 2 * hi;

  v8f acc[2][4] = {};
  wave_gemm_32x64(aptr, 16LL * S_DIM, bptr, 16LL * S_DIM, S_DIM, acc);

  float* eb = energy + (long long)b * D_DIM * D_DIM;
#pragma unroll
  for (int i = 0; i < 2; ++i)
#pragma unroll
    for (int j = 0; j < 4; ++j)
#pragma unroll
      for (int r = 0; r < 8; ++r) {
        const int d = m0 + 16 * i + r + 8 * hi;
        const int e = n0 + 16 * j + lr;
        eb[(long long)d * D_DIM + e] = acc[i][j][r];
      }
}

// ---------------------------------------------------------------------------
// Kernel 3: row softmax over 1024 elements, in place in d_out attn region;
// also writes attnT[b][e][d] to workspace for the final GEMM.
// grid: (1024, 4), block: 1024 (32 waves).
// ---------------------------------------------------------------------------
__global__ void softmax_kernel(float* __restrict__ attn,
                               float* __restrict__ attnT) {
  const int d = blockIdx.x;
  const int b = blockIdx.y;
  const int t = threadIdx.x;            // e index
  float* rowp = attn + ((long long)b * D_DIM + d) * D_DIM;

  __shared__ float redmax[32];
  __shared__ float redsum[32];
  __shared__ float bmax, bsum;

  float x = rowp[t];

  // block max
  float m = x;
#pragma unroll
  for (int o = 16; o > 0; o >>= 1) m = fmaxf(m, __shfl_xor(m, o, 32));
  if ((t & 31) == 0) redmax[t >> 5] = m;
  __syncthreads();
  if (t < 32) {
    float mm = redmax[t];
#pragma unroll
    for (int o = 16; o > 0; o >>= 1) mm = fmaxf(mm, __shfl_xor(mm, o, 32));
    if (t == 0) bmax = mm;
  }
  __syncthreads();
  const float rowmax = bmax;

  const float ex = __expf(x - rowmax);

  // block sum
  float s = ex;
#pragma unroll
  for (int o = 16; o > 0; o >>= 1) s += __shfl_xor(s, o, 32);
  if ((t & 31) == 0) redsum[t >> 5] = s;
  __syncthreads();
  if (t < 32) {
    float ss = redsum[t];
#pragma unroll
    for (int o = 16; o > 0; o >>= 1) ss += __shfl_xor(ss, o, 32);
    if (t == 0) bsum = ss;
  }
  __syncthreads();

  const float a = ex / bsum;
  rowp[t] = a;
  attnT[((long long)b * D_DIM + t) * D_DIM + d] = a;
}

// ---------------------------------------------------------------------------
// Kernel 4: out[b][s][e] = sum_d v[b][s][d] * attnT[b][e][d]
// M=4096, N=1024, K=1024 per batch.  grid: (16, 16, 4), block 256.
// ---------------------------------------------------------------------------
__global__ void out_kernel(const float* __restrict__ vv,
                           const float* __restrict__ attnT,
                           float* __restrict__ out) {
  const int lane = threadIdx.x & 31;
  const int wave = threadIdx.x >> 5;
  const int lr = lane & 15;
  const int hi = lane >> 4;
  const int m0 = blockIdx.x * 256 + wave * 32;   // s
  const int n0 = blockIdx.y * 64;                // e
  const int b  = blockIdx.z;

  const float* vb = vv    + (long long)b * S_DIM * D_DIM;
  const float* tb = attnT + (long long)b * D_DIM * D_DIM;
  const float* aptr = vb + (long long)(m0 + lr) * D_DIM + 2 * hi;
  const float* bptr = tb + (long long)(n0 + lr) * D_DIM + 2 * hi;

  v8f acc[2][4] = {};
  wave_gemm_32x64(aptr, 16LL * D_DIM, bptr, 16LL * D_DIM, D_DIM, acc);

  float* ob = out + (long long)b * S_DIM * D_DIM;
#pragma unroll
  for (int i = 0; i < 2; ++i)
#pragma unroll
    for (int j = 0; j < 4; ++j)
#pragma unroll
      for (int r = 0; r < 8; ++r) {
        const int s = m0 + 16 * i + r + 8 * hi;
        const int e = n0 + 16 * j + lr;
        ob[(long long)s * D_DIM + e] = acc[i][j][r];
      }
}

// ---------------------------------------------------------------------------
extern "C" void kernel_launch(void* const* d_in, const int* in_sizes, int n_in,
                              void* d_out, int out_size, void* d_ws, size_t ws_size,
                              hipStream_t stream) {
  const float* batch = (const float*)d_in[0];
  const float* Wq = (const float*)d_in[1];
  const float* bq = (const float*)d_in[2];
  const float* Wk = (const float*)d_in[3];
  const float* bk = (const float*)d_in[4];
  const float* Wv = (const float*)d_in[5];
  const float* bv = (const float*)d_in[6];

  float* out  = (float*)d_out;                               // [4,4096,1024]
  float* attn = out + (size_t)B_DIM * S_DIM * D_DIM;         // [4,1024,1024]

  float* qT    = (float*)d_ws;                               // [4,1024,4096]
  float* kT    = qT + (size_t)B_DIM * D_DIM * S_DIM;         // [4,1024,4096]
  float* vvb   = kT + (size_t)B_DIM * D_DIM * S_DIM;         // [4,4096,1024]
  float* attnT = vvb + (size_t)B_DIM * S_DIM * D_DIM;        // [4,1024,1024]

  dim3 blk(256);
  proj_qkv_kernel<<<dim3(64, 16, 3), blk, 0, stream>>>(batch, Wq, bq, Wk, bk,
                                                       Wv, bv, qT, kT, vvb);
  energy_kernel<<<dim3(4, 16, 4), blk, 0, stream>>>(qT, kT, attn);
  softmax_kernel<<<dim3(1024, 4), dim3(1024), 0, stream>>>(attn, attnT);
  out_kernel<<<dim3(16, 16, 4), blk, 0, stream>>>(vvb, attnT, out);
}